// Loss_60610578481375
// MI455X (gfx1250) — compile-verified
//
#include <hip/hip_runtime.h>
#include <hip/hip_bf16.h>

typedef float v2f __attribute__((ext_vector_type(2)));
typedef float v8f __attribute__((ext_vector_type(8)));

#define NB 1024   // blocks in pass 1 == rows of partial matrix P (multiple of 4)
#define BS 256    // 8 wave32 per block

// Pointwise terms: mae/mse (per element) and the two relative losses (per point).
__device__ __forceinline__ void acc_point(float d0, float l0, float o0,
                                          float d1, float l1, float o1,
                                          float& s_mae, float& s_mse,
                                          float& s_mre, float& s_mrde)
{
    float e0 = d0 - l0, e1 = d1 - l1;
    s_mae += fabsf(e0) + fabsf(e1);
    float sq = e0 * e0 + e1 * e1;
    s_mse += sq;
    float num  = __builtin_amdgcn_sqrtf(sq);
    float a0 = l0 + o0, a1 = l1 + o1;
    float den1 = __builtin_amdgcn_sqrtf(a0 * a0 + a1 * a1) + 0.01f;       // ||label+v_old|| + eps
    float den2 = __builtin_amdgcn_sqrtf(l0 * l0 + l1 * l1) + 0.01f;       // ||label|| + eps
    s_mre  += num * __builtin_amdgcn_rcpf(den1);
    s_mrde += num * __builtin_amdgcn_rcpf(den2);
}

// Per-face contribution to dH1 - dH2 (up to the global 1e-4*0.5*r scale):
//   net(v,Aup,Adn) = relu(v)*Aup - relu(-v)*Adn,  v = s*u, s>0
//   net1-net2      = s * [ dr*Aup - (dr - du)*Adn ],  dr = relu(u1)-relu(u2), du = u1-u2
__device__ __forceinline__ float dnet(float u1, float du, float Aup, float Adn)
{
    float u2 = u1 - du;
    float dr = fmaxf(u1, 0.0f) - fmaxf(u2, 0.0f);
    return dr * Aup - (dr - du) * Adn;
}

__global__ __launch_bounds__(BS)
void loss_partials(const float* __restrict__ dv, const float* __restrict__ lb,
                   const float* __restrict__ vo, const float* __restrict__ A,
                   float* __restrict__ P)
{
    const int W  = 257;             // point grid
    const int Wc = 256;             // cell grid
    const int plane   = W * W;      // 66049 floats; channel offset (264196 B, fits 24-bit imm)
    const int nstride = 2 * plane;
    const int aplane  = Wc * Wc;
    const int NROWS   = 256 * W;    // 65792 (n,y) rows
    const float kMCE  = 1e-4f * 512.0f;   // vs * 0.5 * r = 1e-4 * 0.5 * 1024

    float s_mae = 0.f, s_mse = 0.f, s_sre = 0.f, s_mre = 0.f, s_mrde = 0.f, s_mce = 0.f;
    const int x = threadIdx.x;

    for (int row = blockIdx.x; row < NROWS; row += NB) {
        // row is wave-uniform -> scalar division, scalar base addresses
        int n = row / W;
        int y = row - n * W;
        const float* b0 = dv + n * nstride + y * W;   // dv, channel 0, row y
        const float* b1 = lb + n * nstride + y * W;   // label
        const float* b2 = vo + n * nstride + y * W;   // v_old

        // own point (y, x)
        float d0 = b0[x],         l0 = b1[x],         o0 = b2[x];
        float d1 = b0[x + plane], l1 = b1[x + plane], o1 = b2[x + plane];
        acc_point(d0, l0, o0, d1, l1, o1, s_mae, s_mse, s_mre, s_mrde);

        // tail point (y, 256) handled by thread 0
        if (x == 0) {
            acc_point(b0[256], b1[256], b2[256],
                      b0[256 + plane], b1[256 + plane], b2[256 + plane],
                      s_mae, s_mse, s_mre, s_mrde);
        }

        if (y < Wc) {   // uniform branch: this row owns a row of cells
            // corner loads: 01 = (y,x+1), 10 = (y+1,x), 11 = (y+1,x+1)
            float d0_01 = b0[x + 1],         d0_10 = b0[x + W],         d0_11 = b0[x + W + 1];
            float l0_01 = b1[x + 1],         l0_10 = b1[x + W],         l0_11 = b1[x + W + 1];
            float o0_01 = b2[x + 1],         o0_10 = b2[x + W],         o0_11 = b2[x + W + 1];
            float d1_01 = b0[x + plane + 1], d1_10 = b0[x + plane + W], d1_11 = b0[x + plane + W + 1];
            float l1_01 = b1[x + plane + 1], l1_10 = b1[x + plane + W], l1_11 = b1[x + plane + W + 1];
            float o1_01 = b2[x + plane + 1], o1_10 = b2[x + plane + W], o1_11 = b2[x + plane + W + 1];

            float e0_00 = d0 - l0,           e1_00 = d1 - l1;
            float e0_01 = d0_01 - l0_01, e0_10 = d0_10 - l0_10, e0_11 = d0_11 - l0_11;
            float e1_01 = d1_01 - l1_01, e1_10 = d1_10 - l1_10, e1_11 = d1_11 - l1_11;

            // ---- strain rate (0.5 folded exactly: 0.25 / 0.125 power-of-two scales) ----
            float rx0 = e0_01 + e0_11 - e0_00 - e0_10;
            float rx1 = e1_01 + e1_11 - e1_00 - e1_10;
            float ry0 = e0_10 + e0_11 - e0_00 - e0_01;
            float ry1 = e1_10 + e1_11 - e1_00 - e1_01;
            float tm  = rx1 + ry0;  // = 2*(ex1 + ey0)
            s_sre += 0.25f * (rx0 * rx0 + ry1 * ry1) + 0.125f * (tm * tm);

            // ---- advection difference ----
            // raw w1 corner velocities (scale folded out): q = d + o
            float q0_00 = d0 + o0, q0_01 = d0_01 + o0_01, q0_10 = d0_10 + o0_10, q0_11 = d0_11 + o0_11;
            float q1_00 = d1 + o1, q1_01 = d1_01 + o1_01, q1_10 = d1_10 + o1_10, q1_11 = d1_11 + o1_11;

            const float* Ab = A + n * aplane + y * Wc;  // A row y
            float Ac = Ab[x];
            float acc = 0.f;
            if (y >= 1)       // north x-face (uniform): dH += fx - gx
                acc += dnet(q0_00 + q0_01, e0_00 + e0_01, Ab[x - Wc], Ac);
            if (y <= Wc - 2)  // south x-face (uniform): dH += gx - fx
                acc -= dnet(q0_10 + q0_11, e0_10 + e0_11, Ac, Ab[x + Wc]);
            if (x >= 1)       // west y-face
                acc += dnet(q1_00 + q1_10, e1_00 + e1_10, Ab[x - 1], Ac);
            if (x <= Wc - 2)  // east y-face
                acc -= dnet(q1_01 + q1_11, e1_01 + e1_11, Ac, Ab[x + 1]);
            float dd = kMCE * acc;
            s_mce += dd * dd;
        }
    }

    // ---- block reduction: wave32 shuffle, then LDS across 8 waves ----
    __shared__ float red[8][6];
    float vals[6] = { s_mae, s_mse, s_sre, s_mre, s_mrde, s_mce };
    #pragma unroll
    for (int m = 0; m < 6; ++m) {
        float v = vals[m];
        #pragma unroll
        for (int off = 16; off > 0; off >>= 1) v += __shfl_down(v, off, 32);
        vals[m] = v;
    }
    int lane = threadIdx.x & 31, wid = threadIdx.x >> 5;
    if (lane == 0) {
        #pragma unroll
        for (int m = 0; m < 6; ++m) red[wid][m] = vals[m];
    }
    __syncthreads();
    if (threadIdx.x == 0) {
        #pragma unroll
        for (int m = 0; m < 6; ++m) {
            float s = 0.f;
            #pragma unroll
            for (int w = 0; w < 8; ++w) s += red[w][m];
            P[blockIdx.x * 16 + m] = s;
        }
        #pragma unroll
        for (int m = 6; m < 16; ++m) P[blockIdx.x * 16 + m] = 0.f;
    }
}

// Final reduction: totals[n] = sum_rows P[row][n] == ones(16x4) x P-chunk(4x16),
// accumulated over NB/4 chunks with V_WMMA_F32_16X16X4_F32 C-chaining.
// A = all-ones (layout independent); only B's N = lane%16 mapping matters since
// the WMMA sums over every K slot and we place 4 distinct rows in the 4 slots.
__global__ __launch_bounds__(32)
void loss_finalize(const float* __restrict__ P, float* __restrict__ out)
{
    int lane = threadIdx.x;
    int col  = lane & 15;
    int half = (lane >> 4) & 1;

    v2f a; a.x = 1.0f; a.y = 1.0f;   // ones 16x4 A-matrix
    v8f c = {};                       // accumulator C/D
    for (int base = 0; base < NB; base += 4) {
        v2f b;
        b.x = P[(base + half) * 16 + col];
        b.y = P[(base + 2 + half) * 16 + col];
        c = __builtin_amdgcn_wmma_f32_16x16x4_f32(
                /*neg_a=*/false, a, /*neg_b=*/false, b,
                /*c_mod=*/(short)0, c, /*reuse_a=*/false, /*reuse_b=*/false);
    }
    // D[m][n] identical over m; VGPR0 of lane L holds (M=0|8, N=L%16)
    float total = c[0];

    const float cntPt2  = 33817088.0f;  // 256*2*257*257 (exact in fp32)
    const float cntPt   = 16908544.0f;  // 256*257*257   (exact in fp32)
    const float cntCell = 16777216.0f;  // 256*256*256 = 2^24
    float mean;
    if (col <= 1)                  mean = total / cntPt2;   // mae, mse
    else if (col == 2 || col == 5) mean = total / cntCell;  // sre, mce
    else                           mean = total / cntPt;    // mre, mrde

    float mae  = __shfl(mean, 0, 32);
    float mse  = __shfl(mean, 1, 32);
    float sre  = __shfl(mean, 2, 32);
    float mre  = __shfl(mean, 3, 32);
    float mrde = __shfl(mean, 4, 32);
    float mce  = __shfl(mean, 5, 32);
    if (lane == 0) {
        out[0] = mae;
        out[1] = mse;
        out[2] = sre;
        out[3] = mse + sre;
        out[4] = mse + 0.1f * mre;
        out[5] = mse + mrde;
        out[6] = mse + sre + mrde;
        out[7] = mre;
        out[8] = mrde;
        out[9] = mce;
    }
}

extern "C" void kernel_launch(void* const* d_in, const int* in_sizes, int n_in,
                              void* d_out, int out_size, void* d_ws, size_t ws_size,
                              hipStream_t stream) {
    const float* dv = (const float*)d_in[0];
    const float* lb = (const float*)d_in[1];
    const float* vo = (const float*)d_in[2];
    const float* A  = (const float*)d_in[3];
    float* out = (float*)d_out;
    float* P   = (float*)d_ws;          // NB x 16 fp32 partials (64 KB)

    loss_partials<<<NB, BS, 0, stream>>>(dv, lb, vo, A, P);
    loss_finalize<<<1, 32, 0, stream>>>(P, out);
}